// QuantumLayer_56607668961323
// MI455X (gfx1250) — compile-verified
//
#include <hip/hip_runtime.h>

// QuantumLayer closed form:
//   out[b,i] = prod_{j<=i} cos(x[b,j]) * cos(w[j])
// (product state after RX/RY gives per-qubit factor cos(w_j)cos(x_j); the
// CNOT chain is a basis permutation whose parity telescopes, so each <Z_i>
// factorizes into a cumulative product).
//
// Memory-bound streaming kernel: 4MB in + 4MB out (~0.34us at 23.3TB/s).
// No matmul structure -> no WMMA. CDNA5 paths used: async global->LDS b128
// loads (ASYNCcnt) covered by independent scalar work, ds_load_b128,
// v_cos_f32 TRANS ops, nontemporal b128 stores.

typedef __attribute__((ext_vector_type(4))) float v4f;

__global__ __launch_bounds__(256) void qlayer_kernel(
    const float* __restrict__ x,
    const float* __restrict__ w,
    float* __restrict__ out,
    int rows) {
  __shared__ v4f tile[256];

  const int tid = threadIdx.x;
  const int gid = blockIdx.x * 256 + tid;
  // Clamp (grid is exact for BATCH=262144; clamp keeps EXEC full around asm)
  const int g = gid < rows ? gid : rows - 1;

  const v4f* __restrict__ x4 = reinterpret_cast<const v4f*>(x);
  v4f* __restrict__ o4 = reinterpret_cast<v4f*>(out);

  // ---- issue async global->LDS load of this lane's row (16B) -------------
  // LDS byte address: generic pointers to __shared__ carry the LDS offset in
  // their low 32 bits (ISA 10.2 aperture rules).
  unsigned lds_addr = (unsigned)(size_t)(&tile[tid]);
  unsigned long long gaddr = (unsigned long long)(size_t)(x4 + g);
  asm volatile("global_load_async_to_lds_b128 %0, %1, off"
               :
               : "v"(lds_addr), "v"(gaddr)
               : "memory");

  // ---- independent uniform work hides the async latency ------------------
  // Weight factors: scalar s_load + cos, identical across lanes.
  float cw0 = __cosf(w[0]);
  float cw1 = __cosf(w[1]);
  float cw2 = __cosf(w[2]);
  float cw3 = __cosf(w[3]);
  float W0 = cw0;
  float W1 = W0 * cw1;
  float W2 = W1 * cw2;
  float W3 = W2 * cw3;

  // ---- wait for async data, read own LDS slot (unpredicated) -------------
#if __has_builtin(__builtin_amdgcn_s_wait_asynccnt)
  __builtin_amdgcn_s_wait_asynccnt(0);
#else
  asm volatile("s_wait_asynccnt 0" ::: "memory");
#endif
  v4f xv = tile[tid];  // ds_load_b128; each lane touches only its own 16B

  // ---- per-row cumulative cosine product (v_cos_f32 TRANS ops) -----------
  float c0 = __cosf(xv.x);
  float c1 = __cosf(xv.y);
  float c2 = __cosf(xv.z);
  float c3 = __cosf(xv.w);
  float p0 = c0;
  float p1 = p0 * c1;
  float p2 = p1 * c2;
  float p3 = p2 * c3;

  v4f ov;
  ov.x = p0 * W0;
  ov.y = p1 * W1;
  ov.z = p2 * W2;
  ov.w = p3 * W3;

  // Only the store is predicated.
  if (gid < rows) {
    __builtin_nontemporal_store(ov, &o4[gid]);  // b128 NT store
  }
}

extern "C" void kernel_launch(void* const* d_in, const int* in_sizes, int n_in,
                              void* d_out, int out_size, void* d_ws, size_t ws_size,
                              hipStream_t stream) {
  const float* x = (const float*)d_in[0];      // (BATCH, 4) float32
  const float* w = (const float*)d_in[1];      // (4,) float32
  float* out = (float*)d_out;                  // (BATCH, 4) float32

  int rows = in_sizes[0] / 4;                  // 262144
  dim3 block(256);
  dim3 grid((rows + 255) / 256);               // 1024 blocks, 8 waves/block
  qlayer_kernel<<<grid, block, 0, stream>>>(x, w, out, rows);
}